// ContractingRENODE_89962384982403
// MI455X (gfx1250) — compile-verified
//
#include <hip/hip_runtime.h>
#include <hip/hip_bf16.h>

// ---------------------------------------------------------------------------
// ContractingRENODE forward for MI455X (gfx1250, wave32, WMMA).
// All heavy matrix work goes through V_WMMA_F32_16X16X4_F32 (fp32 tensor op,
// exact precision match with the fp32 reference). The sequential tanh
// triangular solve runs in LDS, 32 batch rows per workgroup.
// ---------------------------------------------------------------------------

typedef __attribute__((ext_vector_type(2))) float v2f;
typedef __attribute__((ext_vector_type(8))) float v8f;

#define ALPHA_C 0.5f
#define EPS_C   0.01f

#define NB 4096
#define NX 512
#define NY 256
#define NU 256
#define NQ 256
#define NH (NX + NQ)      /* 768  */
#define KC (NX + NQ + NU) /* 1024 : columns of Xc = [x | w | u] */

// ---------------------------------------------------------------------------
// Generic fp32 GEMM: C[M,N] = A[M,K] * B[N,K]^T   (row-major, all f32)
// Requirements (satisfied by every call below): M%128==0, N%128==0, K%4==0,
// lda/ldb even (float2 loads).
// Block = 128 threads = 4 waves, block tile 128x128, wave tile 64x64
// (4x4 accumulators of 16x16).
//
// V_WMMA_F32_16X16X4_F32 per-lane layout (ISA 7.12.2):
//   A (16x4):  lane m16 = lane%16 holds A[m16][k0], A[m16][k0+1], k0 = 2*(lane/16)
//   B (4x16):  WMMA_B[k][n] = Bmem[n][k]  -> lane holds Bmem[n=lane%16][k0..k0+1]
//   Both operands are contiguous float2 loads from row-major memory.
//   C/D (16x16): VGPR g = C[g + 8*(lane/16)][lane%16]
// ---------------------------------------------------------------------------
__global__ __launch_bounds__(128)
void gemm_abt_f32(const float* __restrict__ A, const float* __restrict__ B,
                  float* __restrict__ C, int M, int N, int K,
                  int lda, int ldb, int ldc)
{
    const int lane = threadIdx.x & 31;
    const int wv   = threadIdx.x >> 5;
    const int m16  = lane & 15;
    const int kh   = lane >> 4;   // 0 or 1 -> k0 = 2*kh

    const int waveM = blockIdx.y * 128 + (wv >> 1) * 64;
    const int waveN = blockIdx.x * 128 + (wv & 1) * 64;

    const v8f vzero = {0.f, 0.f, 0.f, 0.f, 0.f, 0.f, 0.f, 0.f};
    v8f acc[4][4];
#pragma unroll
    for (int i = 0; i < 4; ++i)
#pragma unroll
        for (int j = 0; j < 4; ++j) acc[i][j] = vzero;

    const float* Arow[4];
    const float* Brow[4];
#pragma unroll
    for (int i = 0; i < 4; ++i)
        Arow[i] = A + (size_t)(waveM + 16 * i + m16) * lda + 2 * kh;
#pragma unroll
    for (int j = 0; j < 4; ++j)
        Brow[j] = B + (size_t)(waveN + 16 * j + m16) * ldb + 2 * kh;

    for (int kk = 0; kk < K; kk += 4) {
        v2f aF[4], bF[4];
#pragma unroll
        for (int i = 0; i < 4; ++i) aF[i] = *(const v2f*)(Arow[i] + kk);
#pragma unroll
        for (int j = 0; j < 4; ++j) bF[j] = *(const v2f*)(Brow[j] + kk);
#pragma unroll
        for (int i = 0; i < 4; ++i)
#pragma unroll
            for (int j = 0; j < 4; ++j)
                acc[i][j] = __builtin_amdgcn_wmma_f32_16x16x4_f32(
                    false, aF[i], false, bF[j], (short)0, acc[i][j], false, false);
    }

#pragma unroll
    for (int i = 0; i < 4; ++i)
#pragma unroll
        for (int j = 0; j < 4; ++j) {
            float* Cp = C + (size_t)(waveM + 16 * i + 8 * kh) * ldc
                          + (waveN + 16 * j + m16);
#pragma unroll
            for (int g = 0; g < 8; ++g)
                Cp[(size_t)g * ldc] = acc[i][j][g];
        }
}

// ---------------------------------------------------------------------------
// Frame element-wise kernels
// ---------------------------------------------------------------------------

// M1T[n,k] = M1[k,n] = -0.5*(H11[k,n] (+eps if diag)) - (S[k,n]-S[n,k]), 512x512
__global__ void k_frame_m1t(const float* __restrict__ H, const float* __restrict__ S,
                            float* __restrict__ M1T)
{
    int idx = blockIdx.x * blockDim.x + threadIdx.x;
    if (idx >= NX * NX) return;
    int n = idx / NX, k = idx - n * NX;
    float h = H[(size_t)k * NH + n];
    if (n == k) h += EPS_C;
    M1T[(size_t)n * NX + k] = -0.5f * h - (S[(size_t)k * NX + n] - S[(size_t)n * NX + k]);
}

// M2T[n,k] = M2[k,n] = -H12[k,n] - U[k,n], n<256, k<512
__global__ void k_frame_m2t(const float* __restrict__ H, const float* __restrict__ U,
                            float* __restrict__ M2T)
{
    int idx = blockIdx.x * blockDim.x + threadIdx.x;
    if (idx >= NQ * NX) return;
    int n = idx / NX, k = idx - n * NX;
    M2T[(size_t)n * NX + k] = -H[(size_t)k * NH + NX + n] - U[(size_t)k * NQ + n];
}

// lam[k] = 0.5*(H22[k,k] + eps)
__global__ void k_lam(const float* __restrict__ H, float* __restrict__ lam)
{
    int t = threadIdx.x;
    if (t < NQ) lam[t] = 0.5f * (H[(size_t)(NX + t) * NH + NX + t] + EPS_C);
}

// D11[k,j] = (j<k) ? -H22[k,j]/lam[k] : 0
__global__ void k_d11(const float* __restrict__ H, const float* __restrict__ lam,
                      float* __restrict__ D11)
{
    int idx = blockIdx.x * blockDim.x + threadIdx.x;
    if (idx >= NQ * NQ) return;
    int k = idx / NQ, j = idx - k * NQ;
    D11[idx] = (j < k) ? (-H[(size_t)(NX + k) * NH + NX + j] / lam[k]) : 0.0f;
}

// Wa[k, :] = [ C1[k,0:512] = U[:,k]^T/lam[k] | zeros(256) | D12[k,0:256] ]
__global__ void k_build_wa(const float* __restrict__ U, const float* __restrict__ D12,
                           const float* __restrict__ lam, float* __restrict__ Wa)
{
    int idx = blockIdx.x * blockDim.x + threadIdx.x;
    if (idx >= NQ * KC) return;
    int k = idx / KC, i = idx - k * KC;
    float v;
    if (i < NX)            v = U[(size_t)i * NQ + k] / lam[k];
    else if (i < NX + NQ)  v = 0.0f;
    else                   v = D12[(size_t)k * NU + (i - NX - NQ)];
    Wa[idx] = v;
}

// Wy[m, :] = [ C2[m,0:512] | D21[m,0:256] ]
__global__ void k_build_wy(const float* __restrict__ C2, const float* __restrict__ D21,
                           float* __restrict__ Wy)
{
    int idx = blockIdx.x * blockDim.x + threadIdx.x;
    if (idx >= NY * (NX + NQ)) return;
    int m = idx / (NX + NQ), i = idx - m * (NX + NQ);
    Wy[idx] = (i < NX) ? C2[(size_t)m * NX + i] : D21[(size_t)m * NQ + (i - NX)];
}

// After A/B1 GEMMs landed in Wd[:, 0:768]: append B2 and subtract alpha on diag
__global__ void k_wd_rest(const float* __restrict__ B2, float* __restrict__ Wd)
{
    int idx = blockIdx.x * blockDim.x + threadIdx.x;
    if (idx >= NX * NU) return;
    int i = idx / NU, j = idx - i * NU;
    Wd[(size_t)i * KC + (NX + NQ) + j] = B2[(size_t)i * NU + j];
    if (j == 0) Wd[(size_t)i * KC + i] -= ALPHA_C;
}

// Xc[b, :] = [ x[b,0:512] | zeros(256) (w slot) | u[b,0:256] ]
__global__ void k_pack_xc(const float* __restrict__ x, const float* __restrict__ u,
                          float* __restrict__ Xc)
{
    int idx = blockIdx.x * blockDim.x + threadIdx.x;
    if (idx >= NB * KC) return;
    int b = idx >> 10, c = idx & (KC - 1);
    float v;
    if (c < NX)            v = x[(size_t)b * NX + c];
    else if (c < NX + NQ)  v = 0.0f;
    else                   v = u[(size_t)b * NU + (c - NX - NQ)];
    Xc[idx] = v;
}

// ---------------------------------------------------------------------------
// Sequential triangular tanh solve:
//   w[b,k] = tanh( (a[b,k] + sum_{j<k} w[b,j]*D11[k,j]) / lam[k] )
// 32 batch rows per block, w tile lives in LDS, 8 waves each own a 32-wide
// j-chunk; D11[k,j]=0 for j>=k so no masking needed (w tile zero-inited).
// Writes w into Xc columns [NX, NX+NQ).
// ---------------------------------------------------------------------------
#define SR 32
__global__ __launch_bounds__(256)
void solve_w_kernel(const float* __restrict__ aG, const float* __restrict__ D11,
                    const float* __restrict__ lam, float* __restrict__ Xc)
{
    __shared__ float wS[SR][NQ + 1];   // +1 pad: conflict-free row reads
    __shared__ float dRow[NQ];
    __shared__ float lamS[NQ];
    __shared__ float part[8][SR];

    const int t = threadIdx.x;
    const int r = t & 31;    // row within tile (lane)
    const int q = t >> 5;    // wave id = j-chunk 0..7
    const int rowBase = blockIdx.x * SR;

    for (int c = t; c < SR * (NQ + 1); c += 256) ((float*)wS)[c] = 0.0f;
    lamS[t] = lam[t];
    __syncthreads();

    for (int k = 0; k < NQ; ++k) {
        dRow[t] = D11[(size_t)k * NQ + t];
        __syncthreads();

        float s = 0.0f;
        const int jb = q * 32;
#pragma unroll 8
        for (int jj = 0; jj < 32; ++jj)
            s += wS[r][jb + jj] * dRow[jb + jj];
        part[q][r] = s;
        __syncthreads();

        if (t < SR) {
            float tot = aG[(size_t)(rowBase + t) * NQ + k];
#pragma unroll
            for (int qq = 0; qq < 8; ++qq) tot += part[qq][t];
            wS[t][k] = tanhf(tot / lamS[k]);
        }
        __syncthreads();
    }

    // write w tile into Xc's w slot (coalesced)
    for (int idx = t; idx < SR * NQ; idx += 256) {
        int rr = idx >> 8;
        int cc = idx & (NQ - 1);
        Xc[(size_t)(rowBase + rr) * KC + NX + cc] = wS[rr][cc];
    }
}

// ---------------------------------------------------------------------------
// Host-side orchestration
// ---------------------------------------------------------------------------
extern "C" void kernel_launch(void* const* d_in, const int* in_sizes, int n_in,
                              void* d_out, int out_size, void* d_ws, size_t ws_size,
                              hipStream_t stream)
{
    const float* u     = (const float*)d_in[0];
    const float* x     = (const float*)d_in[1];
    const float* Xm    = (const float*)d_in[2];
    const float* S     = (const float*)d_in[3];
    const float* P_inv = (const float*)d_in[4];
    const float* U     = (const float*)d_in[5];
    const float* D12   = (const float*)d_in[6];
    const float* B2    = (const float*)d_in[7];
    const float* C2    = (const float*)d_in[8];
    const float* D21   = (const float*)d_in[9];
    // d_in[10] = D22 (zeros) -> contributes nothing, skipped.

    float* dx_out = (float*)d_out;                       // 4096 x 512
    float* y_out  = dx_out + (size_t)NB * NX;            // 4096 x 256

    // workspace layout (persistent region, then temps aliased under Xc)
    float* w = (float*)d_ws;
    size_t off = 0;
    float* lam = w + off; off += NQ;
    float* D11 = w + off; off += (size_t)NQ * NQ;
    float* Wa  = w + off; off += (size_t)NQ * KC;
    float* Wy  = w + off; off += (size_t)NY * (NX + NQ);
    float* Wd  = w + off; off += (size_t)NX * KC;
    float* tempBase = w + off;
    float* H   = tempBase;                    // 768x768
    float* Pi  = H   + (size_t)NH * NH;       // 512x512
    float* M1T = Pi  + (size_t)NX * NX;       // 512x512
    float* M2T = M1T + (size_t)NX * NX;       // 256x512
    // temps are dead before Xc is packed -> alias:
    float* Xc  = tempBase;                    // 4096x1024
    float* aG  = Xc + (size_t)NB * KC;        // 4096x256

    auto gemm = [&](const float* A, const float* B, float* C,
                    int M, int N, int K, int lda, int ldb, int ldc) {
        dim3 grid(N / 128, M / 128);
        gemm_abt_f32<<<grid, 128, 0, stream>>>(A, B, C, M, N, K, lda, ldb, ldc);
    };

    // ---- frame ----
    gemm(Xm, Xm, H, NH, NH, NH, NH, NH, NH);                         // H = X X^T
    gemm(P_inv, P_inv, Pi, NX, NX, NX, NX, NX, NX);                  // Pi = P P^T
    k_lam<<<1, 256, 0, stream>>>(H, lam);
    k_frame_m1t<<<(NX * NX) / 256, 256, 0, stream>>>(H, S, M1T);
    k_frame_m2t<<<(NQ * NX) / 256, 256, 0, stream>>>(H, U, M2T);
    k_d11<<<(NQ * NQ) / 256, 256, 0, stream>>>(H, lam, D11);
    k_build_wa<<<(NQ * KC) / 256, 256, 0, stream>>>(U, D12, lam, Wa);
    k_build_wy<<<(NY * (NX + NQ)) / 256, 256, 0, stream>>>(C2, D21, Wy);
    gemm(Pi, M1T, Wd,        NX, NX, NX, NX, NX, KC);                // A  -> Wd[:,0:512]
    gemm(Pi, M2T, Wd + NX,   NX, NQ, NX, NX, NX, KC);                // B1 -> Wd[:,512:768]
    k_wd_rest<<<(NX * NU) / 256, 256, 0, stream>>>(B2, Wd);          // B2, -alpha*I

    // ---- batched path ----
    k_pack_xc<<<((size_t)NB * KC) / 256, 256, 0, stream>>>(x, u, Xc);
    gemm(Xc, Wa, aG, NB, NQ, KC, KC, KC, NQ);                        // a
    solve_w_kernel<<<NB / SR, 256, 0, stream>>>(aG, D11, lam, Xc);   // w -> Xc[:,512:768]
    gemm(Xc, Wd, dx_out, NB, NX, KC, KC, KC, NX);                    // dx
    gemm(Xc, Wy, y_out, NB, NQ, NX + NQ, KC, NX + NQ, NQ);           // y
}